// AttentionDecoder_42322607735440
// MI455X (gfx1250) — compile-verified
//
#include <hip/hip_runtime.h>
#include <math.h>

typedef __attribute__((ext_vector_type(2))) float v2f;
typedef __attribute__((ext_vector_type(8))) float v8f;

#define T_DIM 32768
#define H_DIM 1024
#define O_DIM 1024

// workspace layout (in floats)
#define WS_SCORES 0              // 32768 scores
#define WS_STATS  32768          // [0] = max, [1] = sum(exp)
#define WS_PART   32772          // 128 * 1024 partial outputs (16B aligned)
#define NBLK_C    128
#define ROWS_C    256            // rows per block in pass C

// ---------------------------------------------------------------------------
// Pass A: scores[t] = dot(enc[t,:], w_enc)  via V_WMMA_F32_16X16X4_F32.
// Per wave: 16 rows. A-matrix (16x4 f32): only row M=0 nonzero = w_enc[kb..kb+3]
//   -> lane 0 holds {K0,K1}, lane 16 holds {K2,K3}, all other lanes zero.
// B-matrix (4x16 f32): lane L<16 -> N=L holds {K0,K1}; lane L>=16 -> N=L-16
//   holds {K2,K3}.  B[k][n] = enc[rowBase+n][kb+k].
// D row M=0 (VGPR0, lanes 0..15) accumulates the 16 scores.
// ---------------------------------------------------------------------------
__global__ __launch_bounds__(256) void scores_wmma_kernel(
    const float* __restrict__ enc, const float* __restrict__ attn_w,
    float* __restrict__ ws)
{
    const int lane    = threadIdx.x & 31;
    const int waveId  = threadIdx.x >> 5;
    const int rowBase = blockIdx.x * 128 + waveId * 16;
    const int n       = lane & 15;
    const int colPair = (lane >> 4) << 1;          // 0 for lanes 0-15, 2 for 16-31
    const bool isRow0 = (n == 0);

    const float* __restrict__ w_enc = attn_w + O_DIM;       // attn_w[:, O:]
    const float* bptr = enc + (size_t)(rowBase + n) * H_DIM + colPair;
    const float* aptr = w_enc + colPair;

    v8f c = {};
#pragma unroll 4
    for (int kb = 0; kb < H_DIM; kb += 4) {
        v2f b  = *(const v2f*)(bptr + kb);
        v2f wv = *(const v2f*)(aptr + kb);
        v2f a;
        a.x = isRow0 ? wv.x : 0.0f;
        a.y = isRow0 ? wv.y : 0.0f;
        // 8 args: (neg_a, A, neg_b, B, c_mod, C, reuse_a, reuse_b)
        c = __builtin_amdgcn_wmma_f32_16x16x4_f32(
                false, a, false, b, (short)0, c, false, false);
    }
    // D[M=0][N=lane] lives in c[0] of lanes 0..15
    if (lane < 16) ws[WS_SCORES + rowBase + lane] = c[0];
}

// ---------------------------------------------------------------------------
// Pass B: global max and sum(exp(s - max)) over the 32768 scores.
// Single block, fixed-order tree reductions -> deterministic.
// ---------------------------------------------------------------------------
__global__ __launch_bounds__(1024) void softmax_stats_kernel(float* __restrict__ ws)
{
    __shared__ float red[1024];
    const int tid = threadIdx.x;

    float m = -INFINITY;
    for (int i = tid; i < T_DIM; i += 1024) m = fmaxf(m, ws[WS_SCORES + i]);
    red[tid] = m;
    __syncthreads();
    for (int s = 512; s > 0; s >>= 1) {
        if (tid < s) red[tid] = fmaxf(red[tid], red[tid + s]);
        __syncthreads();
    }
    const float gmax = red[0];
    __syncthreads();

    float sum = 0.0f;
    for (int i = tid; i < T_DIM; i += 1024) sum += __expf(ws[WS_SCORES + i] - gmax);
    red[tid] = sum;
    __syncthreads();
    for (int s = 512; s > 0; s >>= 1) {
        if (tid < s) red[tid] += red[tid + s];
        __syncthreads();
    }
    if (tid == 0) { ws[WS_STATS] = gmax; ws[WS_STATS + 1] = red[0]; }
}

// ---------------------------------------------------------------------------
// Pass C: weights -> d_out[1024 + t]; per-block partial Σ_t w[t]*enc[t,:].
// Each thread owns a float4 column slice (fully coalesced 4KB/row reads,
// mostly L2 hits since enc (128 MiB) fits in 192 MB L2 after pass A).
// ---------------------------------------------------------------------------
__global__ __launch_bounds__(256) void weighted_sum_kernel(
    const float* __restrict__ enc, float* __restrict__ ws,
    float* __restrict__ d_out)
{
    __shared__ float wbuf[ROWS_C];
    const int tid = threadIdx.x;
    const int r0  = blockIdx.x * ROWS_C;

    const float gmax    = ws[WS_STATS];
    const float inv_sum = 1.0f / ws[WS_STATS + 1];

    // phase 1: one weight per thread
    {
        const float w = __expf(ws[WS_SCORES + r0 + tid] - gmax) * inv_sum;
        wbuf[tid] = w;
        d_out[H_DIM + r0 + tid] = w;     // normalized_weights output
    }
    __syncthreads();

    const int h = tid * 4;               // 256 threads * 4 cols = 1024
    const float* p = enc + (size_t)r0 * H_DIM + h;
    float4 acc = make_float4(0.f, 0.f, 0.f, 0.f);
    for (int r = 0; r < ROWS_C; ++r) {
        const int rpf = (r + 8 < ROWS_C) ? (r + 8) : (ROWS_C - 1);
        __builtin_prefetch(p + (size_t)rpf * H_DIM, 0, 0);   // global_prefetch_b8
        const float4 e = *(const float4*)(p + (size_t)r * H_DIM);
        const float  w = wbuf[r];
        acc.x += w * e.x; acc.y += w * e.y; acc.z += w * e.z; acc.w += w * e.w;
    }
    *(float4*)(ws + WS_PART + (size_t)blockIdx.x * H_DIM + h) = acc;
}

// ---------------------------------------------------------------------------
// Pass D: deterministic fixed-order combine of the 128 partials.
// ---------------------------------------------------------------------------
__global__ __launch_bounds__(256) void combine_kernel(
    const float* __restrict__ ws, float* __restrict__ d_out)
{
    const int h = blockIdx.x * 256 + threadIdx.x;
    float acc = 0.0f;
    for (int b = 0; b < NBLK_C; ++b)
        acc += ws[WS_PART + (size_t)b * H_DIM + h];
    d_out[h] = acc;                      // output vector (first 1024 of d_out)
}

// ---------------------------------------------------------------------------
extern "C" void kernel_launch(void* const* d_in, const int* in_sizes, int n_in,
                              void* d_out, int out_size, void* d_ws, size_t ws_size,
                              hipStream_t stream)
{
    // inputs: 0=dec_h(1024) 1=enc(32768*1024) 2=attn_w(2048) 3=attn_b(1)
    // dec_h / attn_w[:,:O] / attn_b only shift every score by the same
    // constant; softmax is shift-invariant, so they cannot affect either
    // output -> intentionally unread.
    (void)in_sizes; (void)n_in; (void)out_size; (void)ws_size;
    const float* enc    = (const float*)d_in[1];
    const float* attn_w = (const float*)d_in[2];
    float* ws  = (float*)d_ws;
    float* out = (float*)d_out;

    scores_wmma_kernel <<<T_DIM / 128, 256, 0, stream>>>(enc, attn_w, ws);
    softmax_stats_kernel<<<1, 1024, 0, stream>>>(ws);
    weighted_sum_kernel<<<NBLK_C, 256, 0, stream>>>(enc, ws, out);
    combine_kernel      <<<H_DIM / 256, 256, 0, stream>>>(ws, out);
}